// Decoder_15693810499724
// MI455X (gfx1250) — compile-verified
//
#include <hip/hip_runtime.h>

// CDNA5 / gfx1250, wave32. Fused decoder: (mixture*mask) -> GEMM vs W^T via
// V_WMMA_F32_16X16X4_F32 -> overlap-and-add.
//
// Memory-bound problem (~400MB streamed, 2.1 GFLOP f32): keep f32 end-to-end,
// use f32 WMMA so the N=512 contraction runs on the matrix pipe, stream
// est_mask with non-temporal loads to preserve L2 residency of mixture_w.

typedef float v2f __attribute__((ext_vector_type(2)));
typedef float v8f __attribute__((ext_vector_type(8)));

#define M_DIM 4
#define C_DIM 2
#define N_DIM 512
#define K_DIM 16000
#define L_DIM 16
#define HOP   8
#define T_DIM ((K_DIM - 1) * HOP + L_DIM)   // 128008
#define NTK   (K_DIM / 16)                  // 1000 k-tiles of 16 frames

__global__ void zero_f32(float* __restrict__ p, int n) {
  int i = blockIdx.x * blockDim.x + threadIdx.x;
  if (i < n) p[i] = 0.0f;
}

__global__ __launch_bounds__(256) void decoder_wmma(
    const float* __restrict__ mixture,   // [M, N, K]
    const float* __restrict__ est,       // [M, C, N, K]
    const float* __restrict__ W,         // [L, N]  (torch Linear weight)
    float* __restrict__ out)             // [M, C, T]  (pre-zeroed)
{
  // W^T staged in LDS, pair-interleaved so each WMMA step's B-fragment
  // (n, n+1) is one aligned 8-byte ds_load_b64 per lane (conflict-free:
  // 32 lanes x 2 dwords = 64 banks exactly once).
  //   WtP[p*32 + l*2 + par] = W[l][2p + par]
  __shared__ float WtP[N_DIM * L_DIM];    // 32 KB
  __shared__ float fr[8][16][17];         // per-wave 16x16 D tile, padded

  const int tid  = threadIdx.x;
  const int wid  = tid >> 5;
  const int lane = tid & 31;
  const int kl   = lane & 15;   // A: frame-row M ; B/D: column l
  const int hi   = lane >> 4;   // half-wave select (Kdim pairing 0/2, 1/3)

  for (int i = tid; i < N_DIM * L_DIM; i += 256) {
    int p   = i >> 5;
    int rem = i & 31;
    int l   = rem >> 1;
    int par = rem & 1;
    WtP[i] = W[l * N_DIM + (2 * p + par)];
  }
  __syncthreads();

  // tile = ((m*NTK + ktile)*C + c): c innermost so paired waves share the
  // same mixture tile (L2/temporal reuse of the 131MB mixture tensor).
  const int tile  = blockIdx.x * 8 + wid;
  const int c     = tile & 1;
  const int t2    = tile >> 1;
  const int ktile = t2 % NTK;
  const int m     = t2 / NTK;
  const int k0    = ktile * 16;

  // A-fragment address per ISA layout: VGPR0 = (M=kl, Kdim = 2*hi),
  // VGPR1 = (M=kl, Kdim = 2*hi + 1). A[M=k][Kdim=n] = mixture[n][k]*mask[n][k].
  const float* ep = est     + ((size_t)(m * C_DIM + c) * N_DIM + 2 * hi) * K_DIM + k0 + kl;
  const float* mp = mixture + ((size_t)m * N_DIM + 2 * hi) * K_DIM + k0 + kl;

  v8f acc = {};
#pragma unroll 4
  for (int n0 = 0; n0 < N_DIM; n0 += 4) {
    v2f a;
    // est_mask is single-use: non-temporal (TH=NT) keeps L2 for mixture.
    a.x = __builtin_nontemporal_load(&ep[0])     * mp[0];
    a.y = __builtin_nontemporal_load(&ep[K_DIM]) * mp[K_DIM];
    ep += 4 * (size_t)K_DIM;
    mp += 4 * (size_t)K_DIM;

    // B[Kdim = n0+2*hi (+1)][N = kl] = W[kl][n]: one ds_load_b64.
    v2f b = *(const v2f*)&WtP[((n0 >> 1) + hi) * 32 + kl * 2];

    // D(16x16 f32) += A(16x4 f32) x B(4x16 f32)
    acc = __builtin_amdgcn_wmma_f32_16x16x4_f32(
        /*neg_a=*/false, a, /*neg_b=*/false, b,
        /*c_mod=*/(short)0, acc, /*reuse_a=*/false, /*reuse_b=*/false);
  }

  // D layout: VGPR j -> (M=j, N=lane) lanes 0-15 ; (M=j+8, N=lane-16) hi half.
  // fr[wid][frame_row][l]
#pragma unroll
  for (int j = 0; j < 8; ++j)
    fr[wid][j + 8 * hi][kl] = acc[j];
  // same-wave LDS ops are in-order; fr[wid] is private to this wave.

  // Fused overlap-and-add: tile covers t in [8*k0, 8*k0 + 136).
  // t = 8q + r gets frames[q][r] + frames[q-1][r+8]. Interior samples
  // ([8,128)) are fully owned by this tile -> plain store; the 8-sample
  // seams at each end are shared with the neighbor tile -> atomic add.
  float* op = out + (size_t)(m * C_DIM + c) * T_DIM + (size_t)k0 * HOP;
  for (int s = lane; s < 136; s += 32) {
    int q = s >> 3, r = s & 7;
    float v = 0.0f;
    if (q < 16) v += fr[wid][q][r];
    if (q >= 1) v += fr[wid][q - 1][r + 8];
    if (s >= 8 && s < 128) op[s] = v;
    else                   atomicAdd(&op[s], v);
  }
}

extern "C" void kernel_launch(void* const* d_in, const int* in_sizes, int n_in,
                              void* d_out, int out_size, void* d_ws, size_t ws_size,
                              hipStream_t stream) {
  const float* mixture = (const float*)d_in[0];  // [4, 512, 16000]
  const float* est     = (const float*)d_in[1];  // [4, 2, 512, 16000]
  const float* W       = (const float*)d_in[2];  // [16, 512]
  float* out = (float*)d_out;                    // [4, 2, 128008]

  zero_f32<<<(out_size + 255) / 256, 256, 0, stream>>>(out, out_size);

  const int tiles  = M_DIM * C_DIM * NTK;        // 8000 wave-tiles
  const int blocks = tiles / 8;                  // 8 waves per 256-thread block
  decoder_wmma<<<blocks, 256, 0, stream>>>(mixture, est, W, out);
}